// ISTA_24197845745786
// MI455X (gfx1250) — compile-verified
//
#include <hip/hip_runtime.h>
#include <math.h>

// ISTA on MI455X (gfx1250). fp32 WMMA (V_WMMA_F32_16X16X4_F32) matvecs,
// H (128 MB) is L2-resident (192 MB L2) -> bandwidth-bound, ~2 matvecs/iter.

#define NN 4096
#define MM 8192
#define MAX_ITER 100
#define MU_C  2e-5f
#define RHO_C 0.01f
#define BETA_C (MU_C * RHO_C)
#define EPS_C 1e-6f

typedef __attribute__((ext_vector_type(2))) float v2f;
typedef __attribute__((ext_vector_type(8))) float v8f;

__device__ __forceinline__ v8f wmma_f32_k4(v2f a, v2f b, v8f c) {
    // D = A(16x4 f32) x B(4x16 f32) + C(16x16 f32)
    return __builtin_amdgcn_wmma_f32_16x16x4_f32(
        /*neg_a=*/false, a, /*neg_b=*/false, b,
        /*c_mod=*/(short)0, c, /*reuse_a=*/false, /*reuse_b=*/false);
}

// ---------------------------------------------------------------------------
// init: zero s (in d_out), y, done flag, l1_total
// ---------------------------------------------------------------------------
__global__ void ista_init(float* __restrict__ s, float* __restrict__ y,
                          int* __restrict__ done, float* __restrict__ l1_total) {
    int idx = blockIdx.x * blockDim.x + threadIdx.x;   // 32*256 = 8192 threads
    if (idx < MM) s[idx] = 0.0f;
    if (idx < NN) y[idx] = 0.0f;
    if (idx == 0) { *done = 0; *l1_total = 0.0f; }
}

// ---------------------------------------------------------------------------
// K1: g = s - MU * H^T (y - x);  s_new = soft(g, beta);  s_out = done ? s : s_new
//     16 columns (j-group) per block; 8 waves split K = N = 4096 (512 each).
//     WMMA: A row0 = temp chunk, B = H[k..k+3][j0..j0+15].
// ---------------------------------------------------------------------------
__global__ void ista_k1_update_s(const float* __restrict__ H,
                                 const float* __restrict__ x,
                                 const float* __restrict__ y,
                                 float* __restrict__ s,
                                 const int* __restrict__ done,
                                 float* __restrict__ diff_part,
                                 float* __restrict__ l1_part) {
    __shared__ float temp[NN];         // 16 KB: y - x staged once per block
    __shared__ float red[8][16];
    __shared__ float sred[2][16];

    const int tid  = threadIdx.x;      // 256 threads = 8 waves
    const int wave = tid >> 5;
    const int lane = tid & 31;
    const int j0   = blockIdx.x * 16;  // 512 blocks cover M

    for (int r = tid; r < NN; r += 256) temp[r] = y[r] - x[r];
    __syncthreads();

    const int  kBeg = wave * 512;
    const int  kEnd = kBeg + 512;
    const int  rofs = (lane >> 4) << 1;          // lanes<16 -> 0, lanes>=16 -> 2
    const bool en   = (lane & 15) == 0;          // only A row 0 is nonzero
    const float* Hc = H + (size_t)j0 + (lane & 15);

    v8f acc = {0.f, 0.f, 0.f, 0.f, 0.f, 0.f, 0.f, 0.f};

#pragma unroll 4
    for (int k0 = kBeg; k0 < kEnd; k0 += 4) {
        // A: temp[k0..k0+3] in row 0 (LDS broadcast reads, masked to row 0)
        float t0 = temp[k0 + rofs];
        float t1 = temp[k0 + rofs + 1];
        v2f a; a.x = en ? t0 : 0.0f; a.y = en ? t1 : 0.0f;
        // B: H rows k0+rofs, k0+rofs+1 at columns j0..j0+15 (coalesced 64B/row)
        v2f b;
        b.x = Hc[(size_t)(k0 + rofs)     * MM];
        b.y = Hc[(size_t)(k0 + rofs + 1) * MM];
        __builtin_prefetch(&Hc[(size_t)(k0 + rofs + 32) * MM], 0, 1);
        acc = wmma_f32_k4(a, b, acc);
    }

    // D row 0 lives in VGPR0 across lanes 0..15 (N = lane)
    if (lane < 16) red[wave][lane] = acc[0];
    __syncthreads();

    if (tid < 16) {
        float dot = 0.0f;
        for (int w = 0; w < 8; ++w) dot += red[w][tid];
        const int j  = j0 + tid;
        const float sj = s[j];
        const float g  = sj - MU_C * dot;
        const float ag = fabsf(g) - BETA_C;
        const float s_new = (ag > 0.0f) ? copysignf(ag, g) : 0.0f;
        const float s_out = (*done) ? sj : s_new;   // uses done BEFORE this iter's diff
        s[j] = s_out;
        sred[0][tid] = fabsf(s_new - sj);
        sred[1][tid] = fabsf(s_out);
    }
    __syncthreads();
    if (tid == 0) {
        float d = 0.0f, l = 0.0f;
        for (int t = 0; t < 16; ++t) { d += sred[0][t]; l += sred[1][t]; }
        diff_part[blockIdx.x] = d;
        l1_part[blockIdx.x]   = l;
    }
}

// ---------------------------------------------------------------------------
// Kfin1: fixed-order reduce 512 diff/l1 partials; sticky done |= (diff <= eps)
// ---------------------------------------------------------------------------
__global__ void ista_kfin1(const float* __restrict__ diff_part,
                           const float* __restrict__ l1_part,
                           int* __restrict__ done,
                           float* __restrict__ l1_total) {
    __shared__ float rd[256], rl[256];
    const int t = threadIdx.x;
    rd[t] = diff_part[t] + diff_part[t + 256];
    rl[t] = l1_part[t]   + l1_part[t + 256];
    __syncthreads();
    for (int ofs = 128; ofs > 0; ofs >>= 1) {
        if (t < ofs) { rd[t] += rd[t + ofs]; rl[t] += rl[t + ofs]; }
        __syncthreads();
    }
    if (t == 0) {
        if (rd[0] <= EPS_C) *done = 1;
        *l1_total = rl[0];
    }
}

// ---------------------------------------------------------------------------
// K2: y = H @ s_out; per-block partial of sum((y-x)^2).
//     16 rows (i-group) per block; 8 waves split K = M = 8192 (1024 each).
//     WMMA: A = H[i0..i0+15][k..k+3], B col0 = s chunk.
// ---------------------------------------------------------------------------
__global__ void ista_k2_matvec_err(const float* __restrict__ H,
                                   const float* __restrict__ x,
                                   const float* __restrict__ s,
                                   float* __restrict__ y,
                                   float* __restrict__ sq_part) {
    __shared__ float sl[MM];           // 32 KB: s staged once per block
    __shared__ float red[8][16];
    __shared__ float sqred[16];

    const int tid  = threadIdx.x;      // 256 threads = 8 waves
    const int wave = tid >> 5;
    const int lane = tid & 31;
    const int i0   = blockIdx.x * 16;  // 256 blocks cover N

    for (int r = tid; r < MM; r += 256) sl[r] = s[r];
    __syncthreads();

    const int  kBeg = wave * 1024;
    const int  kEnd = kBeg + 1024;
    const int  rofs = (lane >> 4) << 1;
    const bool en   = (lane & 15) == 0;          // only B column 0 is nonzero
    const float* Hr = H + (size_t)(i0 + (lane & 15)) * MM;

    v8f acc = {0.f, 0.f, 0.f, 0.f, 0.f, 0.f, 0.f, 0.f};

#pragma unroll 4
    for (int k0 = kBeg; k0 < kEnd; k0 += 4) {
        // A: lane m<16 holds H[i0+m][k0], H[i0+m][k0+1]; lanes 16-31 hold K=2,3
        v2f a = *(const v2f*)(Hr + k0 + rofs);
        __builtin_prefetch(Hr + k0 + 128, 0, 1);
        // B: s chunk in column 0 only (LDS broadcast reads, masked)
        float b0 = sl[k0 + rofs];
        float b1 = sl[k0 + rofs + 1];
        v2f b; b.x = en ? b0 : 0.0f; b.y = en ? b1 : 0.0f;
        acc = wmma_f32_k4(a, b, acc);
    }

    // D column 0: lane 0 has rows 0..7 in acc[0..7]; lane 16 has rows 8..15
    if ((lane & 15) == 0) {
        const int half = lane >> 4;
        for (int r = 0; r < 8; ++r) red[wave][half * 8 + r] = acc[r];
    }
    __syncthreads();

    if (tid < 16) {
        float dot = 0.0f;
        for (int w = 0; w < 8; ++w) dot += red[w][tid];
        const int i = i0 + tid;
        y[i] = dot;
        const float e = dot - x[i];
        sqred[tid] = e * e;
    }
    __syncthreads();
    if (tid == 0) {
        float sq = 0.0f;
        for (int t = 0; t < 16; ++t) sq += sqred[t];
        sq_part[blockIdx.x] = sq;
    }
}

// ---------------------------------------------------------------------------
// Kfin2: err[k] = 0.5 * sum((y-x)^2) + RHO * ||s||_1
// ---------------------------------------------------------------------------
__global__ void ista_kfin2(const float* __restrict__ sq_part,
                           const float* __restrict__ l1_total,
                           float* __restrict__ err_out) {
    __shared__ float r[256];
    const int t = threadIdx.x;
    r[t] = sq_part[t];                 // exactly 256 partials
    __syncthreads();
    for (int ofs = 128; ofs > 0; ofs >>= 1) {
        if (t < ofs) r[t] += r[t + ofs];
        __syncthreads();
    }
    if (t == 0) *err_out = 0.5f * r[0] + RHO_C * (*l1_total);
}

// ---------------------------------------------------------------------------
extern "C" void kernel_launch(void* const* d_in, const int* in_sizes, int n_in,
                              void* d_out, int out_size, void* d_ws, size_t ws_size,
                              hipStream_t stream) {
    (void)in_sizes; (void)n_in; (void)out_size; (void)ws_size;
    const float* x = (const float*)d_in[0];        // [N,1]
    const float* H = (const float*)d_in[1];        // [N,M] row-major

    float* s   = (float*)d_out;                    // state lives in output
    float* err = (float*)d_out + MM;               // 100 losses

    float* ws        = (float*)d_ws;
    float* y         = ws;                         // N
    float* diff_part = ws + NN;                    // 512
    float* l1_part   = diff_part + 512;            // 512
    float* sq_part   = l1_part + 512;              // 256
    float* l1_total  = sq_part + 256;              // 1
    int*   done      = (int*)(l1_total + 1);       // 1

    ista_init<<<32, 256, 0, stream>>>(s, y, done, l1_total);

    for (int it = 0; it < MAX_ITER; ++it) {
        ista_k1_update_s<<<MM / 16, 256, 0, stream>>>(H, x, y, s, done,
                                                      diff_part, l1_part);
        ista_kfin1<<<1, 256, 0, stream>>>(diff_part, l1_part, done, l1_total);
        ista_k2_matvec_err<<<NN / 16, 256, 0, stream>>>(H, x, s, y, sq_part);
        ista_kfin2<<<1, 256, 0, stream>>>(sq_part, l1_total, err + it);
    }
}